// GNNBlock_78323023610097
// MI455X (gfx1250) — compile-verified
//
#include <hip/hip_runtime.h>
#include <hip/hip_bf16.h>

// ---------------------------------------------------------------------------
// GNN block for MI455X (gfx1250, wave32).
// fp32 GEMMs on the matrix pipe via V_WMMA_F32_16X16X4_F32, with LDS-staged
// A/B slabs (B transposed so WMMA B-fragments are contiguous ds_load_b64).
// Edge aggregation: float4 gathers + fp32 global atomics. BatchNorm via
// per-column block partials + atomics.
// ---------------------------------------------------------------------------

typedef __attribute__((ext_vector_type(2))) float v2f;
typedef __attribute__((ext_vector_type(8))) float v8f;

#define GD 128   // d_model
#define GFF 512  // dim_feedforward
#define KC 64    // K-slab staged in LDS per iteration

// ---------------- utility kernels ----------------

__global__ void zero_f32(float* __restrict__ p, int n) {
    int i = blockIdx.x * blockDim.x + threadIdx.x;
    if (i < n) p[i] = 0.0f;
}

// deg[dst] += edge_weight  (segment_sum over dst)
__global__ void deg_scatter(const int* __restrict__ ei, const float* __restrict__ ew,
                            float* __restrict__ deg, int E) {
    int e = blockIdx.x * blockDim.x + threadIdx.x;
    if (e >= E) return;
    int dst = ei[E + e];
    atomicAdd(&deg[dst], ew[e]);
}

// dinv[i] = rsqrt(deg[i] + 1)   (in place on deg buffer)
__global__ void dinv_kernel(float* __restrict__ deg, int n) {
    int i = blockIdx.x * blockDim.x + threadIdx.x;
    if (i < n) {
        float d = deg[i] + 1.0f;
        deg[i] = rsqrtf(d);
    }
}

// ---------------- WMMA fp32 GEMM with LDS staging ----------------
// C[M,NC] = A[M,K] @ B[K,NC] (+bias) (+relu) (+residual)
// Block = 256 threads = 8 waves; block tile = 16 rows x 128 cols;
// wave w owns cols [w*16, w*16+16). grid = (M/16, NC/128).
// Per KC=64 slab: sA[16][KC+1] (coalesced float4 stage, shared by all waves),
// sB[128][KC+1] transposed ([col][k]) so the B fragment pair (B[k][c],
// B[k+1][c]) is one contiguous ds_load_b64. Pitch KC+1=65 => 129-float...
// (actually pitch 65 floats, 65 mod 64 = 1) => bank-conflict-free.
// Fragment layouts per CDNA5 ISA 7.12.2 (32-bit A 16x4, B 4x16, C/D 16x16).
__global__ void __launch_bounds__(256)
gemm_wmma_f32(const float* __restrict__ A, const float* __restrict__ B,
              const float* __restrict__ bias, const float* __restrict__ Rsd,
              float* __restrict__ C, int M, int K, int NC, int relu) {
    __shared__ float sA[16 * (KC + 1)];
    __shared__ float sB[128 * (KC + 1)];

    const int tid  = threadIdx.x;
    const int lane = tid & 31;
    const int wave = tid >> 5;                 // 0..7
    const int row0 = blockIdx.x * 16;
    const int col0 = blockIdx.y * 128;
    const int n    = lane & 15;                // local row (A) / col (B,C)
    const int kh   = (lane >> 4) << 1;         // 0 or 2: K sub-pair select

    v8f c = {};
    for (int ks = 0; ks < K; ks += KC) {
        // ---- stage A slab: 16 x KC (1024 floats, one float4 per thread) ----
        {
            int idx = tid << 2;                // 0..1020
            int r   = idx >> 6;                // / KC
            int kk  = idx & (KC - 1);
            const float4 v4 = *(const float4*)(A + (long long)(row0 + r) * K + ks + kk);
            float* s = sA + r * (KC + 1) + kk;
            s[0] = v4.x; s[1] = v4.y; s[2] = v4.z; s[3] = v4.w;
        }
        // ---- stage B slab transposed: global [KC][128] -> sB[col][k] ----
        for (int j = tid; j < KC * 128; j += 256) {
            int kk = j >> 7;                   // 0..KC-1
            int cc = j & 127;                  // 0..127
            sB[cc * (KC + 1) + kk] = B[(long long)(ks + kk) * NC + col0 + cc];
        }
        __syncthreads();

        const float* aP = sA + n * (KC + 1) + kh;
        const float* bP = sB + (wave * 16 + n) * (KC + 1) + kh;
#pragma unroll
        for (int k = 0; k < KC; k += 4) {
            v2f a, b;
            a.x = aP[k];
            a.y = aP[k + 1];
            b.x = bP[k];
            b.y = bP[k + 1];
            c = __builtin_amdgcn_wmma_f32_16x16x4_f32(
                    /*neg_a=*/false, a, /*neg_b=*/false, b,
                    /*c_mod=*/(short)0, c, /*reuse_a=*/false, /*reuse_b=*/false);
        }
        __syncthreads();
    }

    const int col = col0 + wave * 16 + n;
    const float bv = bias ? bias[col] : 0.0f;
    const int rbase = row0 + ((lane >> 4) << 3);  // lanes 16-31 hold M=8..15
#pragma unroll
    for (int r = 0; r < 8; ++r) {
        long long idx = (long long)(rbase + r) * NC + col;
        float v = c[r] + bv;
        if (relu) v = fmaxf(v, 0.0f);
        if (Rsd)  v += Rsd[idx];
        C[idx] = v;
    }
}

// ---------------- edge aggregation ----------------
// y[dst,:] += dinv[src]*w*dinv[dst] * h[src,:]
// 32 threads per edge, each handles 4 consecutive floats (float4 gather,
// 4 fp32 global atomics).
__global__ void edge_agg(const int* __restrict__ ei, const float* __restrict__ ew,
                         const float* __restrict__ dinv, const float* __restrict__ h,
                         float* __restrict__ y, int E) {
    int tid = blockIdx.x * blockDim.x + threadIdx.x;
    int e = tid >> 5;
    if (e >= E) return;
    int g = (tid & 31) << 2;                            // 0,4,...,124
    int src = ei[e];
    int dst = ei[E + e];
    float norm = dinv[src] * ew[e] * dinv[dst];
    const float4 hv = *(const float4*)(h + (long long)src * GD + g);
    float* yd = y + (long long)dst * GD + g;
    atomicAdd(yd + 0, norm * hv.x);
    atomicAdd(yd + 1, norm * hv.y);
    atomicAdd(yd + 2, norm * hv.z);
    atomicAdd(yd + 3, norm * hv.w);
}

// y[i,d] += x[i,d] + dinv[i]^2 * h[i,d] + b[d]
__global__ void fuse_add(const float* __restrict__ x, const float* __restrict__ h,
                         const float* __restrict__ dinv, const float* __restrict__ b,
                         float* __restrict__ y, int total) {
    int tid = blockIdx.x * blockDim.x + threadIdx.x;
    if (tid >= total) return;
    int i = tid >> 7;        // /128
    int d = tid & 127;
    float di = dinv[i];
    y[tid] += x[tid] + di * di * h[tid] + b[d];
}

// ---------------- BatchNorm over node axis ----------------
// Per-column partial sums over 256-row stripes, accumulated with atomics.
__global__ void col_stats(const float* __restrict__ y, float* __restrict__ sum,
                          float* __restrict__ sumsq, int N) {
    const int d = threadIdx.x;                          // 0..127
    const int r0 = blockIdx.x * 256;
    float s = 0.0f, s2 = 0.0f;
    for (int r = 0; r < 256; ++r) {
        int row = r0 + r;
        if (row < N) {
            float v = y[(long long)row * GD + d];
            s += v;
            s2 += v * v;
        }
    }
    atomicAdd(&sum[d], s);
    atomicAdd(&sumsq[d], s2);
}

__global__ void bn_finalize(const float* __restrict__ sum, const float* __restrict__ sumsq,
                            float* __restrict__ mu, float* __restrict__ rstd, int N) {
    int d = threadIdx.x;
    if (d >= GD) return;
    float m = sum[d] / (float)N;
    float var = fmaxf(sumsq[d] / (float)N - m * m, 0.0f);
    mu[d] = m;
    rstd[d] = rsqrtf(var + 1e-5f);
}

__global__ void bn_apply(const float* __restrict__ y, const float* __restrict__ mu,
                         const float* __restrict__ rstd, const float* __restrict__ gamma,
                         const float* __restrict__ beta, float* __restrict__ out, int total) {
    int tid = blockIdx.x * blockDim.x + threadIdx.x;
    if (tid >= total) return;
    int d = tid & 127;
    out[tid] = gamma[d] * (y[tid] - mu[d]) * rstd[d] + beta[d];
}

// ---------------- host orchestration ----------------

extern "C" void kernel_launch(void* const* d_in, const int* in_sizes, int n_in,
                              void* d_out, int out_size, void* d_ws, size_t ws_size,
                              hipStream_t stream) {
    const int N = 100000, E = 1600000, D = GD, FF = GFF;

    const float* x     = (const float*)d_in[0];
    const int*   ei    = (const int*)  d_in[1];   // [2,E]: row0=src, row1=dst
    const float* ew    = (const float*)d_in[2];
    const float* W     = (const float*)d_in[3];
    const float* b     = (const float*)d_in[4];
    const float* gamma = (const float*)d_in[5];
    const float* beta  = (const float*)d_in[6];
    const float* W1    = (const float*)d_in[7];
    const float* b1    = (const float*)d_in[8];
    const float* W2    = (const float*)d_in[9];
    const float* b2    = (const float*)d_in[10];
    float* out = (float*)d_out;

    // workspace layout (floats): h | y | x1 | t | dinv | sum | sumsq | mu | rstd
    float* ws   = (float*)d_ws;
    float* h    = ws;                          // N*D
    float* y    = h  + (size_t)N * D;          // N*D (agg accumulator, then y1/y2)
    float* x1   = y  + (size_t)N * D;          // N*D
    float* t    = x1 + (size_t)N * D;          // N*FF
    float* dinv = t  + (size_t)N * FF;         // N   (deg computed here in place)
    float* sum   = dinv + N;                   // D
    float* sumsq = sum + D;                    // D
    float* mu    = sumsq + D;                  // D
    float* rstd  = mu + D;                     // D

    const int ND = N * D;
    const int TB = 256;
    dim3 blk(TB);

    // ---- zero accumulators ----
    zero_f32<<<dim3((ND + TB - 1) / TB), blk, 0, stream>>>(y, ND);
    zero_f32<<<dim3((N + TB - 1) / TB), blk, 0, stream>>>(dinv, N);
    zero_f32<<<dim3(1), blk, 0, stream>>>(sum, 2 * D);   // sum + sumsq contiguous

    // ---- degree -> dinv ----
    deg_scatter<<<dim3((E + TB - 1) / TB), blk, 0, stream>>>(ei, ew, dinv, E);
    dinv_kernel<<<dim3((N + TB - 1) / TB), blk, 0, stream>>>(dinv, N);

    // ---- h = x @ W  (no bias here; b added in fuse step per reference) ----
    gemm_wmma_f32<<<dim3(N / 16, D / 128), blk, 0, stream>>>(
        x, W, nullptr, nullptr, h, N, D, D, 0);

    // ---- edge aggregation into y ----
    {
        long long threads = (long long)E * 32;
        edge_agg<<<dim3((unsigned)((threads + TB - 1) / TB)), blk, 0, stream>>>(
            ei, ew, dinv, h, y, E);
    }

    // ---- y = y + x + dinv^2*h + b ----
    fuse_add<<<dim3((ND + TB - 1) / TB), blk, 0, stream>>>(x, h, dinv, b, y, ND);

    // ---- BN #1 -> x1 ----
    col_stats<<<dim3((N + 255) / 256), dim3(D), 0, stream>>>(y, sum, sumsq, N);
    bn_finalize<<<dim3(1), dim3(D), 0, stream>>>(sum, sumsq, mu, rstd, N);
    bn_apply<<<dim3((ND + TB - 1) / TB), blk, 0, stream>>>(y, mu, rstd, gamma, beta, x1, ND);

    // ---- MLP: t = relu(x1 @ W1 + b1) ----
    gemm_wmma_f32<<<dim3(N / 16, FF / 128), blk, 0, stream>>>(
        x1, W1, b1, nullptr, t, N, D, FF, 1);

    // ---- y2 = t @ W2 + b2 + x1 ----
    gemm_wmma_f32<<<dim3(N / 16, D / 128), blk, 0, stream>>>(
        t, W2, b2, x1, y, N, FF, D, 0);

    // ---- BN #2 -> out ----
    zero_f32<<<dim3(1), blk, 0, stream>>>(sum, 2 * D);
    col_stats<<<dim3((N + 255) / 256), dim3(D), 0, stream>>>(y, sum, sumsq, N);
    bn_finalize<<<dim3(1), dim3(D), 0, stream>>>(sum, sumsq, mu, rstd, N);
    bn_apply<<<dim3((ND + TB - 1) / TB), blk, 0, stream>>>(y, mu, rstd, gamma, beta, out, ND);
}